// PMAT_24842090840470
// MI455X (gfx1250) — compile-verified
//
#include <hip/hip_runtime.h>
#include <hip/hip_bf16.h>
#include <math.h>

typedef __attribute__((ext_vector_type(2))) float v2f;
typedef __attribute__((ext_vector_type(8))) float v8f;

#define DFEAT 128  // reference fixes D=128

__device__ __forceinline__ float selu_f(float x) {
    const float scale = 1.0507009873554805f;
    const float alpha = 1.6732632423543772f;
    return x > 0.f ? scale * x : scale * alpha * (expf(x) - 1.f);
}

__device__ __forceinline__ float sigmoid_f(float x) {
    return 1.f / (1.f + expf(-x));
}

// out[node] = l2norm(in[node] + sigma * noise[node]); one wave32 per node.
__global__ void pmat_l2norm_kernel(const float* __restrict__ in,
                                   const float* __restrict__ noise,
                                   float sigma,
                                   float* __restrict__ out, int N) {
    int node = (int)((blockIdx.x * blockDim.x + threadIdx.x) >> 5);
    int lane = (int)(threadIdx.x & 31u);
    if (node >= N) return;  // uniform per wave
    size_t base = (size_t)node * DFEAT + (size_t)lane * 4;
    float4 v = *(const float4*)(in + base);
    float4 nz = *(const float4*)(noise + base);
    v.x += sigma * nz.x; v.y += sigma * nz.y;
    v.z += sigma * nz.z; v.w += sigma * nz.w;
    float ss = v.x * v.x + v.y * v.y + v.z * v.z + v.w * v.w;
    // wave32 butterfly reduction
    for (int m = 16; m >= 1; m >>= 1) ss += __shfl_xor(ss, m, 32);
    float nrm = sqrtf(ss);
    float s = 1.f / fmaxf(nrm, 1e-12f);
    float4 o; o.x = v.x * s; o.y = v.y * s; o.z = v.z * s; o.w = v.w * s;
    *(float4*)(out + base) = o;
}

// Per-node projections s1[n]=h[n].w1, s2[n]=h[n].w2 via V_WMMA_F32_16X16X4_F32.
// A (16x4) = weight rows: M=0 -> w1, M=1 -> w2, rows 2..15 zero (value-masked,
// loads are address-clamped so no EXEC divergence).
// B (4x16) = h tile transposed: column N = node m0+N, K looped over 128/4 chunks.
// C row M=0 = s1 for 16 nodes (c[0], lanes 0-15), row M=1 = s2 (c[1], lanes 0-15).
__global__ void pmat_proj_wmma_kernel(const float* __restrict__ h,
                                      const float* __restrict__ w,  // [w1(128) | w2(128)]
                                      float* __restrict__ s1,
                                      float* __restrict__ s2, int N) {
    int lane = (int)(threadIdx.x & 31u);
    int wave = (int)(threadIdx.x >> 5);
    int m0 = ((int)blockIdx.x * (int)(blockDim.x >> 5) + wave) * 16;
    if (m0 >= N) return;  // uniform per wave

    int sub = lane & 15;
    int khalf = (lane >> 4) * 2;  // lanes 0-15: K sub 0,1 ; lanes 16-31: K sub 2,3

    // A operand: weight row 'sub' (only rows 0,1 nonzero). Clamp address, mask value.
    const float* wrow = w + (size_t)((sub < 2) ? sub : 0) * DFEAT;
    float amask = (sub < 2) ? 1.f : 0.f;

    // B operand: node column = m0 + sub (clamped to stay in bounds, EXEC all-ones).
    int node = m0 + sub;
    if (node >= N) node = N - 1;
    const float* hrow = h + (size_t)node * DFEAT;

    v8f c = {};
    #pragma unroll 4
    for (int kb = 0; kb < DFEAT / 4; ++kb) {
        int k0 = kb * 4;
        v2f a;
        a.x = wrow[k0 + khalf] * amask;
        a.y = wrow[k0 + khalf + 1] * amask;
        v2f b;
        b.x = hrow[k0 + khalf];
        b.y = hrow[k0 + khalf + 1];
        // 8 args: (neg_a, A, neg_b, B, c_mod, C, reuse_a, reuse_b)
        c = __builtin_amdgcn_wmma_f32_16x16x4_f32(false, a, false, b,
                                                  (short)0, c, false, false);
    }
    // C layout: VGPR v, lanes 0-15 -> (M=v, N=lane). Row 0 = s1, row 1 = s2.
    if (lane < 16 && (m0 + sub) < N) {
        s1[m0 + sub] = c[0];
        s2[m0 + sub] = c[1];
    }
}

// One wave32 per edge: lane 0 computes the uniform scalars and broadcasts;
// all lanes gather a float4 of h[src] and atomic-f32 scatter into aggr[dst].
__global__ void pmat_edge_kernel(const float* __restrict__ h,
                                 const int* __restrict__ src,
                                 const int* __restrict__ dst,
                                 const float* __restrict__ s1,
                                 const float* __restrict__ s2,
                                 const float* __restrict__ bptr,
                                 float* __restrict__ aggr, int E) {
    int eid = (int)((blockIdx.x * blockDim.x + threadIdx.x) >> 5);
    int lane = (int)(threadIdx.x & 31u);
    if (eid >= E) return;  // uniform per wave
    int s = 0, d = 0;
    float alpha = 0.f;
    if (lane == 0) {
        s = src[eid];
        d = dst[eid];
        float e = s1[s] + s2[d] + bptr[0];
        alpha = sigmoid_f(selu_f(e));
        // speculative prefetch of the streaming edge-index arrays
        __builtin_prefetch(src + eid + 2048, 0, 0);
        __builtin_prefetch(dst + eid + 2048, 0, 0);
    }
    s = __shfl(s, 0, 32);
    d = __shfl(d, 0, 32);
    alpha = __shfl(alpha, 0, 32);

    const float4* hp = (const float4*)(h + (size_t)s * DFEAT);
    float4 v = hp[lane];
    float* ap = aggr + (size_t)d * DFEAT + (size_t)lane * 4;
    atomicAdd(ap + 0, alpha * v.x);
    atomicAdd(ap + 1, alpha * v.y);
    atomicAdd(ap + 2, alpha * v.z);
    atomicAdd(ap + 3, alpha * v.w);
}

extern "C" void kernel_launch(void* const* d_in, const int* in_sizes, int n_in,
                              void* d_out, int out_size, void* d_ws, size_t ws_size,
                              hipStream_t stream) {
    const float* x      = (const float*)d_in[0];
    const int*   eidx   = (const int*)d_in[1];
    const float* attn_w = (const float*)d_in[2];
    const float* attn_b = (const float*)d_in[3];
    const float* noise  = (const float*)d_in[4];

    int HOPS = in_sizes[3];
    int D2   = in_sizes[2] / HOPS;   // 2*D
    int D    = D2 / 2;               // = DFEAT = 128
    int N    = in_sizes[0] / D;
    int E    = in_sizes[1] / 2;
    (void)n_in; (void)out_size; (void)ws_size; (void)D;

    float* out = (float*)d_out;
    float* s1 = (float*)d_ws;
    float* s2 = s1 + N;
    const int* src = eidx;        // edge_index row 0
    const int* dst = eidx + E;    // edge_index row 1
    size_t slice = (size_t)N * DFEAT;

    // out[0] = l2norm(x)   (noise=x with sigma=0 -> identity add)
    pmat_l2norm_kernel<<<(N + 7) / 8, 256, 0, stream>>>(x, x, 0.f, out, N);

    for (int k = 0; k < HOPS; ++k) {
        const float* h  = out + (size_t)k * slice;
        float*       hn = out + (size_t)(k + 1) * slice;
        // per-node projections via WMMA (f32 matrix pipe); 8 tiles (waves) per block
        int tiles = (N + 15) / 16;
        pmat_proj_wmma_kernel<<<(tiles + 7) / 8, 256, 0, stream>>>(
            h, attn_w + (size_t)k * D2, s1, s2, N);
        // zero accumulation slice (capture-safe async memset)
        hipMemsetAsync(hn, 0, slice * sizeof(float), stream);
        // edge scatter with f32 global atomics
        pmat_edge_kernel<<<(E + 7) / 8, 256, 0, stream>>>(
            h, src, dst, s1, s2, attn_b + k, hn, E);
        // hn = l2norm(hn + 0.1*noise[k])
        pmat_l2norm_kernel<<<(N + 7) / 8, 256, 0, stream>>>(
            hn, noise + (size_t)k * slice, 0.1f, hn, N);
    }
}